// SparseAttentionDistance_59287728554028
// MI455X (gfx1250) — compile-verified
//
#include <hip/hip_runtime.h>

#define BATCHES 8
#define NPTS    4096
#define DIM     64
#define NBINS   64
#define BINSZ   64      // NPTS / NBINS (equal-size split per reference)
#define NHALF   32      // NBINS / 2
#define TOPK    16
#define LDSPAD  2       // pad rows to 66 floats to spread LDS banks

typedef __attribute__((ext_vector_type(2))) float v2f;
typedef __attribute__((ext_vector_type(8))) float v8f;

// ---------------------------------------------------------------------------
// Kernel 1: zero the 536 MB dense output (bandwidth-dominant step).
// ---------------------------------------------------------------------------
__global__ void zero_out_kernel(float4* __restrict__ out, size_t n4) {
    size_t i      = (size_t)blockIdx.x * blockDim.x + threadIdx.x;
    size_t stride = (size_t)gridDim.x * blockDim.x;
    float4 z = make_float4(0.f, 0.f, 0.f, 0.f);
    for (; i < n4; i += stride) out[i] = z;
}

// ---------------------------------------------------------------------------
// Kernel 2: LSH bin id per point. argmax over [P@R, -(P@R)] with first-max
// tie-break (strict >, ascending index) to match jnp.argmax.
// ---------------------------------------------------------------------------
__global__ void bin_kernel(const float* __restrict__ pts,
                           const float* __restrict__ rot,
                           int* __restrict__ bin_idx) {
    __shared__ float srot[DIM][NHALF];
    int t = threadIdx.x;
    for (int i = t; i < DIM * NHALF; i += blockDim.x)
        ((float*)srot)[i] = rot[i];
    __syncthreads();

    int p = blockIdx.x * blockDim.x + t;          // global point id
    const float* src = pts + (size_t)p * DIM;

    float mul[NHALF];
#pragma unroll
    for (int j = 0; j < NHALF; ++j) mul[j] = 0.f;

    for (int k = 0; k < DIM; k += 4) {
        float4 v = *(const float4*)(src + k);
#pragma unroll
        for (int j = 0; j < NHALF; ++j) {
            mul[j] = fmaf(v.x, srot[k + 0][j],
                     fmaf(v.y, srot[k + 1][j],
                     fmaf(v.z, srot[k + 2][j],
                     fmaf(v.w, srot[k + 3][j], mul[j]))));
        }
    }

    float best = mul[0];
    int   bi   = 0;
#pragma unroll
    for (int j = 1; j < NHALF; ++j)
        if (mul[j] > best) { best = mul[j]; bi = j; }
#pragma unroll
    for (int j = 0; j < NHALF; ++j) {
        float c = -mul[j];
        if (c > best) { best = c; bi = NHALF + j; }
    }
    bin_idx[p] = bi;
}

// ---------------------------------------------------------------------------
// Kernel 3: per-batch STABLE counting sort of point ids by bin id.
// Thread b owns bin b; scans indices ascending -> stable, deterministic.
// ---------------------------------------------------------------------------
__global__ void sort_kernel(const int* __restrict__ bin_idx,
                            int* __restrict__ order) {
    __shared__ int sbin[NPTS];
    __shared__ int scnt[NBINS];
    int b     = threadIdx.x;       // 0..63 == bin id
    int batch = blockIdx.x;
    const int* src = bin_idx + batch * NPTS;

    for (int i = b; i < NPTS; i += NBINS) sbin[i] = src[i];
    __syncthreads();

    int cnt = 0;
    for (int i = 0; i < NPTS; ++i) cnt += (sbin[i] == b);
    scnt[b] = cnt;
    __syncthreads();

    int start = 0;
    for (int j = 0; j < b; ++j) start += scnt[j];

    int* dst = order + batch * NPTS;
    int  pos = start;
    for (int i = 0; i < NPTS; ++i)
        if (sbin[i] == b) dst[pos++] = i;
}

// ---------------------------------------------------------------------------
// Kernel 4: per (batch, bin): gather+normalize, S = N*N^T via fp32 WMMA,
// row softmax, exact top-16 selection, scatter into dense output.
// 128 threads = 4 waves; wave w computes rows [16w, 16w+16) of the 64x64 S.
// ---------------------------------------------------------------------------
__global__ __launch_bounds__(128)
void bin_attn_kernel(const float* __restrict__ pts,
                     const int* __restrict__ order,
                     float* __restrict__ out) {
    __shared__ int   sids[BINSZ];
    __shared__ float sp[BINSZ][DIM + LDSPAD];   // normalized points
    __shared__ float ss[BINSZ][DIM + LDSPAD];   // softmax probabilities

    int t     = threadIdx.x;
    int batch = blockIdx.x >> 6;
    int bin   = blockIdx.x & 63;

    // ---- gather + l2-normalize (tf.nn.l2_normalize semantics) ----
    if (t < BINSZ) {
        int id = order[batch * NPTS + bin * BINSZ + t];
        sids[t] = id;
        const float* src = pts + ((size_t)batch * NPTS + id) * DIM;
        float sq = 0.f;
#pragma unroll
        for (int k = 0; k < DIM; k += 4) {
            float4 v = *(const float4*)(src + k);
            sp[t][k + 0] = v.x; sp[t][k + 1] = v.y;
            sp[t][k + 2] = v.z; sp[t][k + 3] = v.w;
            sq = fmaf(v.x, v.x, fmaf(v.y, v.y, fmaf(v.z, v.z, fmaf(v.w, v.w, sq))));
        }
        float scale = rsqrtf(fmaxf(sq, 1e-12f));
#pragma unroll
        for (int k = 0; k < DIM; ++k) sp[t][k] *= scale;
    }
    __syncthreads();

    // ---- S = P * P^T with V_WMMA_F32_16X16X4_F32 ----
    // A 16x4 f32 layout: lanes 0-15 hold K={0,1} (V0,V1), lanes 16-31 K={2,3}.
    // B 4x16 layout is symmetric (B = A^T here), so both fragments are the
    // float2 at sp[rowbase + lane%16][4*kk + 2*(lane/16)].
    int lane = t & 31;
    int wv   = t >> 5;            // wave id -> output row tile
    int lh   = lane & 15;
    int half = lane >> 4;

    v8f acc0 = {}, acc1 = {}, acc2 = {}, acc3 = {};
    const int arow = 16 * wv + lh;
#pragma unroll
    for (int kk = 0; kk < 16; ++kk) {
        int k0 = 4 * kk + 2 * half;
        v2f a  = *(const v2f*)&sp[arow   ][k0];
        v2f b0 = *(const v2f*)&sp[ 0 + lh][k0];
        v2f b1 = *(const v2f*)&sp[16 + lh][k0];
        v2f b2 = *(const v2f*)&sp[32 + lh][k0];
        v2f b3 = *(const v2f*)&sp[48 + lh][k0];
        acc0 = __builtin_amdgcn_wmma_f32_16x16x4_f32(false, a, false, b0, (short)0, acc0, false, false);
        acc1 = __builtin_amdgcn_wmma_f32_16x16x4_f32(false, a, false, b1, (short)0, acc1, false, false);
        acc2 = __builtin_amdgcn_wmma_f32_16x16x4_f32(false, a, false, b2, (short)0, acc2, false, false);
        acc3 = __builtin_amdgcn_wmma_f32_16x16x4_f32(false, a, false, b3, (short)0, acc3, false, false);
    }

    // ---- row softmax; each row lives in 16 lanes of one half-wave.
    // xor masks 1..8 never cross bit 4, so butterflies stay inside the half.
#pragma unroll
    for (int r = 0; r < 8; ++r) {
        float v0 = acc0[r], v1 = acc1[r], v2 = acc2[r], v3 = acc3[r];
        float m = fmaxf(fmaxf(v0, v1), fmaxf(v2, v3));
#pragma unroll
        for (int off = 8; off >= 1; off >>= 1)
            m = fmaxf(m, __shfl_xor(m, off, 32));
        float e0 = expf(v0 - m), e1 = expf(v1 - m);
        float e2 = expf(v2 - m), e3 = expf(v3 - m);
        float s = e0 + e1 + e2 + e3;
#pragma unroll
        for (int off = 8; off >= 1; off >>= 1)
            s += __shfl_xor(s, off, 32);
        float inv = 1.0f / s;
        int row = 16 * wv + r + 8 * half;   // C layout: VGPR r -> M=r / M=r+8
        ss[row][lh +  0] = e0 * inv;
        ss[row][lh + 16] = e1 * inv;
        ss[row][lh + 32] = e2 * inv;
        ss[row][lh + 48] = e3 * inv;
    }
    __syncthreads();

    // ---- exact top-16 per row: rank = #{strictly greater} + #{equal, lower
    // index}. Exactly 16 elements get rank < 16 (total order), matching
    // top_k's lower-index tie-break. Order irrelevant: values scatter into a
    // dense matrix and (row, col) pairs are unique across the whole batch.
    if (t < BINSZ) {
        int grow = sids[t];
        float* orow = out + ((size_t)batch * NPTS + grow) * (size_t)NPTS;
        for (int e = 0; e < BINSZ; ++e) {
            float v = ss[t][e];
            int rank = 0;
            for (int f = 0; f < BINSZ; ++f) {
                float w = ss[t][f];
                rank += (int)((w > v) | ((w == v) & (f < e)));
            }
            if (rank < TOPK) orow[sids[e]] = v;
        }
    }
}

// ---------------------------------------------------------------------------
extern "C" void kernel_launch(void* const* d_in, const int* in_sizes, int n_in,
                              void* d_out, int out_size, void* d_ws, size_t ws_size,
                              hipStream_t stream) {
    const float* pts = (const float*)d_in[0];   // [8, 4096, 64] f32
    const float* rot = (const float*)d_in[1];   // [64, 32] f32
    float* out = (float*)d_out;                 // [8, 4096, 4096] f32

    int* bin_idx = (int*)d_ws;                  // [8*4096]
    int* order   = bin_idx + BATCHES * NPTS;    // [8*4096]

    size_t n4 = (size_t)out_size / 4;           // out_size divisible by 4
    zero_out_kernel<<<4096, 256, 0, stream>>>((float4*)out, n4);
    bin_kernel<<<(BATCHES * NPTS) / 256, 256, 0, stream>>>(pts, rot, bin_idx);
    sort_kernel<<<BATCHES, NBINS, 0, stream>>>(bin_idx, order);
    bin_attn_kernel<<<BATCHES * NBINS, 128, 0, stream>>>(pts, order, out);
}